// SimpleQuantumModel_69226282877370
// MI455X (gfx1250) — compile-verified
//
#include <hip/hip_runtime.h>

#define NQ 8
#define BATCHSZ 8192
#define PI_F 3.14159265358979323846f

typedef __attribute__((ext_vector_type(2))) float v2f;
typedef __attribute__((ext_vector_type(8))) float v8f;

// ---------------- Quantum statevector kernel: one wave32 per sample ----------------
// Amplitude index = (lane << 3) | r. Qubits 0..2 live in register bits r,
// qubits 3..7 live in lane bits. State: re[8], im[8] per lane (16 VGPRs).

__device__ __forceinline__ void apply_ry(float re[8], float im[8], int q, float th, int lane) {
    float c = __cosf(0.5f * th);
    float s = __sinf(0.5f * th);
    if (q < 3) {
        const int m = 1 << q;
#pragma unroll
        for (int r = 0; r < 8; ++r) {
            if ((r & m) == 0) {
                const int r1 = r | m;
                float a0 = re[r], a1 = re[r1];
                re[r]  = c * a0 - s * a1;
                re[r1] = s * a0 + c * a1;
                float b0 = im[r], b1 = im[r1];
                im[r]  = c * b0 - s * b1;
                im[r1] = s * b0 + c * b1;
            }
        }
    } else {
        const int lm = 1 << (q - 3);
        const float ss = (lane & lm) ? s : -s;  // upper: c*me + s*partner ; lower: c*me - s*partner
#pragma unroll
        for (int r = 0; r < 8; ++r) {
            float pr = __shfl_xor(re[r], lm, 32);
            float pi = __shfl_xor(im[r], lm, 32);
            re[r] = c * re[r] + ss * pr;
            im[r] = c * im[r] + ss * pi;
        }
    }
}

__device__ __forceinline__ void apply_rz(float re[8], float im[8], int q, float th, int lane) {
    float cp = __cosf(0.5f * th);
    float sp = __sinf(0.5f * th);
    if (q < 3) {
        const int m = 1 << q;
#pragma unroll
        for (int r = 0; r < 8; ++r) {
            const float sg = (r & m) ? sp : -sp;   // bit1: e^{+i t/2}; bit0: e^{-i t/2}
            float a = re[r], b = im[r];
            re[r] = a * cp - sg * b;
            im[r] = b * cp + sg * a;
        }
    } else {
        const int lm = 1 << (q - 3);
        const float sg = (lane & lm) ? sp : -sp;
#pragma unroll
        for (int r = 0; r < 8; ++r) {
            float a = re[r], b = im[r];
            re[r] = a * cp - sg * b;
            im[r] = b * cp + sg * a;
        }
    }
}

__device__ __forceinline__ void apply_cnot(float re[8], float im[8], int c, int t, int lane) {
    if (c < 3 && t < 3) {                       // both register bits
        const int cm = 1 << c, tm = 1 << t;
        float nr[8], ni[8];
#pragma unroll
        for (int r = 0; r < 8; ++r) {
            const int src = (r & cm) ? (r ^ tm) : r;
            nr[r] = re[src]; ni[r] = im[src];
        }
#pragma unroll
        for (int r = 0; r < 8; ++r) { re[r] = nr[r]; im[r] = ni[r]; }
    } else if (c < 3 && t >= 3) {               // control in register, target across lanes
        const int cm = 1 << c, lm = 1 << (t - 3);
#pragma unroll
        for (int r = 0; r < 8; ++r) {
            float pr = __shfl_xor(re[r], lm, 32);
            float pi = __shfl_xor(im[r], lm, 32);
            if (r & cm) { re[r] = pr; im[r] = pi; }
        }
    } else if (c >= 3 && t >= 3) {              // both lane bits
        const int lmc = 1 << (c - 3), lmt = 1 << (t - 3);
        const bool ctrl = (lane & lmc) != 0;
#pragma unroll
        for (int r = 0; r < 8; ++r) {
            float pr = __shfl_xor(re[r], lmt, 32);
            float pi = __shfl_xor(im[r], lmt, 32);
            if (ctrl) { re[r] = pr; im[r] = pi; }
        }
    } else {                                    // control in lane bits, target in register
        const int lmc = 1 << (c - 3), tm = 1 << t;
        const bool ctrl = (lane & lmc) != 0;
        float nr[8], ni[8];
#pragma unroll
        for (int r = 0; r < 8; ++r) {
            const int src = ctrl ? (r ^ tm) : r;
            nr[r] = re[src]; ni[r] = im[src];
        }
#pragma unroll
        for (int r = 0; r < 8; ++r) { re[r] = nr[r]; im[r] = ni[r]; }
    }
}

__global__ __launch_bounds__(256) void quantum_kernel(const float* __restrict__ x,
                                                      const float* __restrict__ qw,
                                                      float* __restrict__ zout) {
    const int lane = threadIdx.x & 31;
    const int wave = threadIdx.x >> 5;
    const int s    = blockIdx.x * 8 + wave;

    // normalized encoding angles (uniform across the wave; each lane redundantly computes)
    float xv[NQ];
#pragma unroll
    for (int q = 0; q < NQ; ++q) xv[q] = x[s * NQ + q];
    float xmin = xv[0], xmax = xv[0];
#pragma unroll
    for (int q = 1; q < NQ; ++q) { xmin = fminf(xmin, xv[q]); xmax = fmaxf(xmax, xv[q]); }
    const float inv = 1.0f / (xmax - xmin + 1e-8f);

    // |0...0>
    float re[8], im[8];
#pragma unroll
    for (int r = 0; r < 8; ++r) { re[r] = 0.0f; im[r] = 0.0f; }
    if (lane == 0) re[0] = 1.0f;

    // feature encoding RY(pi * xn)
#pragma unroll
    for (int q = 0; q < NQ; ++q)
        apply_ry(re, im, q, (xv[q] - xmin) * inv * PI_F, lane);

    // variational layers
#pragma unroll
    for (int layer = 0; layer < 3; ++layer) {
#pragma unroll
        for (int q = 0; q < NQ; ++q) {
            apply_ry(re, im, q, qw[layer * 16 + q * 2 + 0], lane);
            apply_rz(re, im, q, qw[layer * 16 + q * 2 + 1], lane);
        }
#pragma unroll
        for (int q = 0; q < NQ; ++q)
            apply_cnot(re, im, q, (q + 1) & 7, lane);
    }

    // <Z_q> expectations
    float p[8], psum = 0.0f;
#pragma unroll
    for (int r = 0; r < 8; ++r) { p[r] = re[r] * re[r] + im[r] * im[r]; psum += p[r]; }

    float zq[NQ];
#pragma unroll
    for (int q = 0; q < 3; ++q) {
        const int m = 1 << q;
        float acc = 0.0f;
#pragma unroll
        for (int r = 0; r < 8; ++r) acc += (r & m) ? -p[r] : p[r];
        zq[q] = acc;
    }
#pragma unroll
    for (int q = 3; q < NQ; ++q)
        zq[q] = (lane & (1 << (q - 3))) ? -psum : psum;

    // wave butterfly reduction
#pragma unroll
    for (int off = 16; off >= 1; off >>= 1) {
#pragma unroll
        for (int q = 0; q < NQ; ++q)
            zq[q] += __shfl_xor(zq[q], off, 32);
    }
    if (lane == 0) {
#pragma unroll
        for (int q = 0; q < NQ; ++q) zout[s * NQ + q] = zq[q];
    }
}

// ---------------- MLP kernel: f32 WMMA 16x16x4 ----------------
// Block = 128 threads (4 waves) handles 16 batch rows.
// Layer1: wave w computes h-tile cols [16w,16w+16), K=8 -> 2 wmma.
// Layer2: wave w computes out-tiles n = 4w..4w+3, K=64 -> 16 wmma each.

__global__ __launch_bounds__(128) void mlp_kernel(const float* __restrict__ z,
                                                  const float* __restrict__ W1,
                                                  const float* __restrict__ b1,
                                                  const float* __restrict__ W2,
                                                  const float* __restrict__ b2,
                                                  float* __restrict__ out) {
    __shared__ float hbuf[16 * 68];   // 16x64 h tile, padded stride 68 (conflict-free)

    const int lane  = threadIdx.x & 31;
    const int wave  = threadIdx.x >> 5;   // 0..3
    const int row0  = blockIdx.x * 16;
    const int mrow  = lane & 15;          // M (or N) index within tile
    const int khalf = lane >> 4;          // 0 or 1: which K-pair this lane holds

    // ---- layer 1 ----
    {
        const int col = wave * 16 + mrow;        // 0..63
        const float bv = b1[col];
        v8f acc = {bv, bv, bv, bv, bv, bv, bv, bv};
#pragma unroll
        for (int ks = 0; ks < 2; ++ks) {
            const int kb = ks * 4;
            v2f a, b;
            a.x = z[(row0 + mrow) * 8 + kb + 2 * khalf + 0];
            a.y = z[(row0 + mrow) * 8 + kb + 2 * khalf + 1];
            b.x = W1[(kb + 2 * khalf + 0) * 64 + col];
            b.y = W1[(kb + 2 * khalf + 1) * 64 + col];
            acc = __builtin_amdgcn_wmma_f32_16x16x4_f32(false, a, false, b,
                                                        (short)0, acc, false, false);
        }
        // relu + stage into LDS; C row M = v + 8*khalf
#pragma unroll
        for (int v = 0; v < 8; ++v) {
            const int m = v + 8 * khalf;
            float h = acc[v];
            hbuf[m * 68 + col] = h > 0.0f ? h : 0.0f;
        }
    }
    __syncthreads();

    // ---- layer 2 ----
#pragma unroll
    for (int nt = 0; nt < 4; ++nt) {
        const int col = (wave * 4 + nt) * 16 + mrow;   // 0..255
        const float bv = b2[col];
        v8f acc = {bv, bv, bv, bv, bv, bv, bv, bv};
#pragma unroll
        for (int ks = 0; ks < 16; ++ks) {
            const int kb = ks * 4;
            v2f a, b;
            a.x = hbuf[mrow * 68 + kb + 2 * khalf + 0];
            a.y = hbuf[mrow * 68 + kb + 2 * khalf + 1];
            b.x = W2[(kb + 2 * khalf + 0) * 256 + col];
            b.y = W2[(kb + 2 * khalf + 1) * 256 + col];
            acc = __builtin_amdgcn_wmma_f32_16x16x4_f32(false, a, false, b,
                                                        (short)0, acc, false, false);
        }
#pragma unroll
        for (int v = 0; v < 8; ++v) {
            const int m = v + 8 * khalf;
            out[(row0 + m) * 256 + col] = acc[v];
        }
    }
}

extern "C" void kernel_launch(void* const* d_in, const int* in_sizes, int n_in,
                              void* d_out, int out_size, void* d_ws, size_t ws_size,
                              hipStream_t stream) {
    const float* x  = (const float*)d_in[0];  // [8192, 8]
    const float* qw = (const float*)d_in[1];  // [3, 8, 2]
    const float* W1 = (const float*)d_in[2];  // [8, 64]
    const float* b1 = (const float*)d_in[3];  // [64]
    const float* W2 = (const float*)d_in[4];  // [64, 256]
    const float* b2 = (const float*)d_in[5];  // [256]
    float* out  = (float*)d_out;              // [8192, 256]
    float* zbuf = (float*)d_ws;               // [8192, 8] staging (256 KB)

    quantum_kernel<<<BATCHSZ / 8, 256, 0, stream>>>(x, qw, zbuf);
    mlp_kernel<<<BATCHSZ / 16, 128, 0, stream>>>(zbuf, W1, b1, W2, b2, out);
}